// ModelNew_29497835389796
// MI455X (gfx1250) — compile-verified
//
#include <hip/hip_runtime.h>
#include <hip/hip_bf16.h>

// ---------------------------------------------------------------------------
// Problem: out[b] = sum_o( x[b,:]·W[o,:] + bias[o] )   (all trailing ops are
// identities on a size-1 axis).  Algebraic collapse:
//     out[b] = x[b,:] · wsum  +  bsum,   wsum[i] = sum_o W[o,i],  bsum = sum bias
// Memory-bound: 256 MB (W) + 32 MB (X) @ 23.3 TB/s  =>  ~12.5 us floor.
// Pipeline (all deterministic, no float atomics):
//   K1: partial column sums of W   (coalesced b128 streaming, 64-way row split)
//   K2: reduce partials -> wsum
//   K3: bias sum -> bsum
//   K4: GEMV  X·wsum via V_WMMA_F32_16X16X4_F32, 8-way K split -> partials
//   K5: reduce K partials + bsum -> out
// ---------------------------------------------------------------------------

typedef __attribute__((ext_vector_type(2))) float v2f;
typedef __attribute__((ext_vector_type(4))) float v4f;
typedef __attribute__((ext_vector_type(8))) float v8f;

#define IN_F   8192
#define OUT_F  8192
#define BATCH  1024

#define ROW_CHUNKS      64                    // K1 row split
#define ROWS_PER_CHUNK  (OUT_F / ROW_CHUNKS)  // 128
#define KSPLIT          8                     // K4 K split
#define K_PER_BLOCK     (IN_F / KSPLIT)       // 1024

// --------------------------- K1: W column partial sums ----------------------
// grid (8, 64), block 256.  Thread t handles 4 consecutive columns -> b128
// loads, lane-consecutive addresses (fully coalesced).  Streams 256 MB once.
__global__ void k_colsum_partial(const float* __restrict__ W,
                                 float* __restrict__ partial) {
    const int col4 = blockIdx.x * 1024 + threadIdx.x * 4;
    const int r0   = blockIdx.y * ROWS_PER_CHUNK;
    v4f acc = {0.f, 0.f, 0.f, 0.f};
    const float* p = W + (size_t)r0 * IN_F + col4;
    #pragma unroll 4
    for (int r = 0; r < ROWS_PER_CHUNK; ++r) {
        acc += *(const v4f*)p;
        p += IN_F;
    }
    *(v4f*)(partial + (size_t)blockIdx.y * IN_F + col4) = acc;
}

// --------------------------- K2: reduce partials -> wsum --------------------
// grid 32, block 256.  Reads 2 MB.
__global__ void k_wsum_reduce(const float* __restrict__ partial,
                              float* __restrict__ wsum) {
    const int col = blockIdx.x * 256 + threadIdx.x;
    float s = 0.f;
    #pragma unroll 8
    for (int c = 0; c < ROW_CHUNKS; ++c)
        s += partial[(size_t)c * IN_F + col];
    wsum[col] = s;
}

// --------------------------- K3: bias sum -----------------------------------
// 1 block, 256 threads, LDS tree reduction (fixed order -> deterministic).
__global__ void k_bias_sum(const float* __restrict__ bias,
                           float* __restrict__ bsum) {
    __shared__ float red[256];
    float s = 0.f;
    for (int i = threadIdx.x; i < OUT_F; i += 256)
        s += bias[i];
    red[threadIdx.x] = s;
    __syncthreads();
    for (int off = 128; off > 0; off >>= 1) {
        if (threadIdx.x < off) red[threadIdx.x] += red[threadIdx.x + off];
        __syncthreads();
    }
    if (threadIdx.x == 0) bsum[0] = red[0];
}

// --------------------------- K4: GEMV via WMMA f32 16x16x4 ------------------
// grid (BATCH/16, KSPLIT), block 32 (one wave32 per block; EXEC all-ones in
// the WMMA loop as required).
//   A (16x4 f32): lane m=lane%16 holds rows of X; K pair = 2*(lane/16)+{0,1}
//   B (4x16 f32): lane n=lane%16 column; we replicate wsum across all 16
//                 columns so every column of D equals the desired dot product.
//   D (16x16 f32 in v8f): column N=0 lives in lane 0 (M=0..7) / lane 16
//                 (M=8..15) -> those two lanes emit the 16 partial outputs.
__global__ void k_gemv_wmma(const float* __restrict__ X,
                            const float* __restrict__ wsum,
                            float* __restrict__ gpart) {
    const int lane  = threadIdx.x;
    const int m     = lane & 15;
    const int koff  = (lane >> 4) * 2;   // 0 for lanes 0-15, 2 for lanes 16-31
    const int rbase = blockIdx.x * 16;
    const int kbeg  = blockIdx.y * K_PER_BLOCK;
    const int kend  = kbeg + K_PER_BLOCK;

    const float* xrow = X + (size_t)(rbase + m) * IN_F;

    v8f c = {};
    for (int k = kbeg; k < kend; k += 4) {
        v2f a = *(const v2f*)(xrow + k + koff);   // A: X[rbase+m][k+koff .. +1]
        v2f b = *(const v2f*)(wsum + k + koff);   // B: wsum replicated over N
        c = __builtin_amdgcn_wmma_f32_16x16x4_f32(
                /*neg_a=*/false, a, /*neg_b=*/false, b,
                /*c_mod=*/(short)0, c, /*reuse_a=*/false, /*reuse_b=*/false);
    }

    // Column N=0 of D: lane 0 -> rows 0..7, lane 16 -> rows 8..15.
    if (m == 0) {
        const int mbase = (lane >> 4) * 8;
        #pragma unroll
        for (int v = 0; v < 8; ++v)
            gpart[(size_t)blockIdx.y * BATCH + rbase + mbase + v] = c[v];
    }
}

// --------------------------- K5: finalize -----------------------------------
__global__ void k_finalize(const float* __restrict__ gpart,
                           const float* __restrict__ bsum,
                           float* __restrict__ out) {
    const int b = blockIdx.x * 256 + threadIdx.x;
    float s = bsum[0];
    #pragma unroll
    for (int c = 0; c < KSPLIT; ++c)
        s += gpart[(size_t)c * BATCH + b];
    out[b] = s;
}

// ---------------------------------------------------------------------------
extern "C" void kernel_launch(void* const* d_in, const int* in_sizes, int n_in,
                              void* d_out, int out_size, void* d_ws, size_t ws_size,
                              hipStream_t stream) {
    const float* x    = (const float*)d_in[0];   // (1024, 8192)
    const float* W    = (const float*)d_in[1];   // (8192, 8192)
    const float* bias = (const float*)d_in[2];   // (8192,)
    float* out = (float*)d_out;                  // (1024, 1)

    // Workspace layout (floats): partials | wsum | bsum(+pad) | gemv partials
    float* ws      = (float*)d_ws;
    float* partial = ws;                                   // 64*8192 = 2 MB
    float* wsum    = partial + (size_t)ROW_CHUNKS * IN_F;  // 8192
    float* bsum    = wsum + IN_F;                          // 1 (padded to 16)
    float* gpart   = bsum + 16;                            // KSPLIT*BATCH

    k_colsum_partial<<<dim3(8, ROW_CHUNKS), 256, 0, stream>>>(W, partial);
    k_wsum_reduce  <<<IN_F / 256, 256, 0, stream>>>(partial, wsum);
    k_bias_sum     <<<1, 256, 0, stream>>>(bias, bsum);
    k_gemv_wmma    <<<dim3(BATCH / 16, KSPLIT), 32, 0, stream>>>(x, wsum, gpart);
    k_finalize     <<<BATCH / 256, 256, 0, stream>>>(gpart, bsum, out);
}